// Attention_70978629533918
// MI455X (gfx1250) — compile-verified
//
#include <hip/hip_runtime.h>
#include <math.h>

typedef float v2f __attribute__((ext_vector_type(2)));
typedef float v8f __attribute__((ext_vector_type(8)));
typedef int   v4i __attribute__((vector_size(16)));   /* matches builtin param type */

typedef __attribute__((address_space(1))) v4i* gvec_t;
typedef __attribute__((address_space(3))) v4i* lvec_t;

#define HIDN 512
#define EMB  2048
#define LSEQ 196
#define BATCH 256
#define NROW (BATCH * LSEQ)   /* 50176 = 784 * 64 */

/* ------------------------------------------------------------------ */
/* Kernel A: Q_h[b,h] = hidden[b,:] . Wq[h,:] + bq[h]                 */
/* ------------------------------------------------------------------ */
__global__ __launch_bounds__(256) void qh_kernel(
    const float* __restrict__ hidden, const float* __restrict__ Wq,
    const float* __restrict__ bq, float* __restrict__ Qh)
{
    __shared__ float hs[HIDN];
    const int b = blockIdx.x, t = threadIdx.x;
    hs[t]       = hidden[b * HIDN + t];
    hs[t + 256] = hidden[b * HIDN + 256 + t];
    __syncthreads();
#pragma unroll
    for (int rep = 0; rep < 2; ++rep) {
        const int h = t + rep * 256;
        float acc = bq[h];
        const float4* w = (const float4*)(Wq + (size_t)h * HIDN);
#pragma unroll 4
        for (int k = 0; k < HIDN / 4; ++k) {
            float4 wv = w[k];
            acc += wv.x * hs[4 * k]     + wv.y * hs[4 * k + 1]
                 + wv.z * hs[4 * k + 2] + wv.w * hs[4 * k + 3];
        }
        Qh[b * HIDN + h] = acc;
    }
}

/* ------------------------------------------------------------------ */
/* Kernel B: fused  K_s = img@Wk^T (+bk +Qh) -> tanh -> dot(Wv) -> e  */
/* 784 WGs x 256 threads (8 waves). Each WG: 64 rows x full H=512.    */
/* A tiles double-buffered in LDS via async global->LDS copies        */
/* (GLOBAL_LOAD_ASYNC_TO_LDS_B128 + s_wait_asynccnt), overlapped with */
/* V_WMMA_F32_16X16X4_F32 compute on the other buffer.                */
/* ------------------------------------------------------------------ */
#define ALDS_STRIDE 68   /* 64 + 4 pad: conflict-free ds_load_b64 */
#define NCHUNK (EMB / 64)

__global__ __launch_bounds__(256) void att_gemm_kernel(
    const float* __restrict__ img,  const float* __restrict__ Wk,
    const float* __restrict__ bk,   const float* __restrict__ Qh,
    const float* __restrict__ Wv,   const float* __restrict__ bv,
    float* __restrict__ e_out)
{
    __shared__ float a_lds[2][64 * ALDS_STRIDE];

    const int mbase = blockIdx.x * 64;
    const int tid   = threadIdx.x;
    const int lane  = tid & 31;
    const int wv_id = tid >> 5;          /* wave 0..7 */
    const int hlf   = lane >> 4;         /* half-wave: selects K pair / row+8 */
    const int lcol  = lane & 15;         /* N (B,C,D) or M (A) lane index */
    const int n0    = wv_id * 64;

    /* B-operand row pointers: B[k][n] = Wk[n][k]; lane=n, hlf picks k+{0,1} vs k+{2,3} */
    const float* bp[4];
#pragma unroll
    for (int nt = 0; nt < 4; ++nt)
        bp[nt] = Wk + (size_t)(n0 + nt * 16 + lcol) * EMB + 2 * hlf;

    /* A fragment LDS offsets per M-subtile */
    int aoff[4];
#pragma unroll
    for (int m = 0; m < 4; ++m)
        aoff[m] = (m * 16 + lcol) * ALDS_STRIDE + 2 * hlf;

    v8f acc[4][4] = {};

    /* async A staging: thread -> (row, 16-col segment), 4x b128 per chunk */
    const int srow = tid >> 2;
    const int sseg = (tid & 3) * 16;
    const float* gsrc = img + (size_t)(mbase + srow) * EMB + sseg;
    float* ldst[2] = { &a_lds[0][srow * ALDS_STRIDE + sseg],
                       &a_lds[1][srow * ALDS_STRIDE + sseg] };

    /* prologue: kick off chunk 0 into buffer 0 */
#pragma unroll
    for (int i = 0; i < 4; ++i)
        __builtin_amdgcn_global_load_async_to_lds_b128(
            (gvec_t)(gsrc + 4 * i), (lvec_t)(ldst[0] + 4 * i), 0, 0);

    for (int c = 0; c < NCHUNK; ++c) {
        const int k0  = c * 64;
        const int cur = c & 1;

        if (c + 1 < NCHUNK) {
            /* prefetch next chunk into the other buffer (safe: trailing
               barrier of chunk c-1 means nobody still reads it) */
            const float* s = gsrc + (c + 1) * 64;
            float*       d = ldst[cur ^ 1];
#pragma unroll
            for (int i = 0; i < 4; ++i)
                __builtin_amdgcn_global_load_async_to_lds_b128(
                    (gvec_t)(s + 4 * i), (lvec_t)(d + 4 * i), 0, 0);
            /* older 4 copies (chunk c) complete; 4 new ones may remain */
            asm volatile("s_wait_asynccnt 0x4" ::: "memory");
        } else {
            asm volatile("s_wait_asynccnt 0x0" ::: "memory");
        }
        __syncthreads();   /* all waves' chunk-c data resident in LDS */

        const float* abase = &a_lds[cur][0];
#pragma unroll
        for (int kk = 0; kk < 64; kk += 4) {
            v2f afr[4], bfr[4];
#pragma unroll
            for (int m = 0; m < 4; ++m)
                afr[m] = *(const v2f*)&abase[aoff[m] + kk];
#pragma unroll
            for (int nt = 0; nt < 4; ++nt)
                bfr[nt] = *(const v2f*)(bp[nt] + k0 + kk);
#pragma unroll
            for (int m = 0; m < 4; ++m)
#pragma unroll
                for (int nt = 0; nt < 4; ++nt)
                    acc[m][nt] = __builtin_amdgcn_wmma_f32_16x16x4_f32(
                        false, afr[m], false, bfr[nt],
                        (short)0, acc[m][nt], false, false);
        }
        __syncthreads();   /* buffer cur free for chunk c+2's async writes */
    }

    /* --- epilogue: tanh + Wv dot, reduce to e[row] --- */
    float wvv[4], bkc[4];
#pragma unroll
    for (int nt = 0; nt < 4; ++nt) {
        const int col = n0 + nt * 16 + lcol;
        wvv[nt] = Wv[col];
        bkc[nt] = bk[col];
    }

    float* psum = &a_lds[0][0];   /* reuse: [64 rows][8 waves] */

#pragma unroll
    for (int m = 0; m < 4; ++m) {
#pragma unroll
        for (int j = 0; j < 8; ++j) {
            const int rloc = m * 16 + j + 8 * hlf;
            const int row  = mbase + rloc;
            const int bb   = row / LSEQ;
            const float* qrow = Qh + (size_t)bb * HIDN;
            float s = 0.f;
#pragma unroll
            for (int nt = 0; nt < 4; ++nt) {
                const int col = n0 + nt * 16 + lcol;
                const float val = acc[m][nt][j] + bkc[nt] + qrow[col];
                s += wvv[nt] * tanhf(val);
            }
            /* reduce across the 16 lanes of this half-wave (rows differ per half) */
            s += __shfl_xor(s, 1);
            s += __shfl_xor(s, 2);
            s += __shfl_xor(s, 4);
            s += __shfl_xor(s, 8);
            if (lcol == 0)
                psum[rloc * 8 + wv_id] = s;
        }
    }
    __syncthreads();

    if (tid < 64) {
        float s = 0.f;
#pragma unroll
        for (int w = 0; w < 8; ++w) s += psum[tid * 8 + w];
        e_out[mbase + tid] = s + bv[0];
    }
}

/* ------------------------------------------------------------------ */
/* Kernel C: softmax over L per batch                                 */
/* ------------------------------------------------------------------ */
__global__ __launch_bounds__(256) void softmax_kernel(
    const float* __restrict__ e, float* __restrict__ alpha)
{
    __shared__ float red[256];
    const int b = blockIdx.x, t = threadIdx.x;
    const float v = (t < LSEQ) ? e[b * LSEQ + t] : -INFINITY;
    red[t] = v;
    __syncthreads();
#pragma unroll
    for (int s = 128; s > 0; s >>= 1) {
        if (t < s) red[t] = fmaxf(red[t], red[t + s]);
        __syncthreads();
    }
    const float mx = red[0];
    __syncthreads();
    const float ex = (t < LSEQ) ? expf(v - mx) : 0.f;
    red[t] = ex;
    __syncthreads();
#pragma unroll
    for (int s = 128; s > 0; s >>= 1) {
        if (t < s) red[t] += red[t + s];
        __syncthreads();
    }
    const float inv = 1.f / red[0];
    if (t < LSEQ) alpha[b * LSEQ + t] = ex * inv;
}

/* ------------------------------------------------------------------ */
/* Kernel D: context[b,e] = sum_l alpha[b,l] * img[b,l,e]             */
/* grid = B*2; each block: one b, 1024 contiguous e (float4/thread)   */
/* ------------------------------------------------------------------ */
__global__ __launch_bounds__(256) void context_kernel(
    const float* __restrict__ img, const float* __restrict__ alpha,
    float* __restrict__ ctx)
{
    __shared__ float al[LSEQ];
    const int b  = blockIdx.x >> 1;
    const int eh = blockIdx.x & 1;
    const int t  = threadIdx.x;
    if (t < LSEQ) al[t] = alpha[b * LSEQ + t];
    __syncthreads();

    const int e0 = eh * 1024 + t * 4;
    const float4* base = (const float4*)(img + (size_t)b * LSEQ * EMB + e0);
    float4 acc = make_float4(0.f, 0.f, 0.f, 0.f);
#pragma unroll 4
    for (int l = 0; l < LSEQ; ++l) {
        const float4 v = base[(size_t)l * (EMB / 4)];
        const float a  = al[l];
        acc.x += a * v.x; acc.y += a * v.y; acc.z += a * v.z; acc.w += a * v.w;
    }
    *(float4*)(ctx + (size_t)b * EMB + e0) = acc;
}

/* ------------------------------------------------------------------ */
extern "C" void kernel_launch(void* const* d_in, const int* in_sizes, int n_in,
                              void* d_out, int out_size, void* d_ws, size_t ws_size,
                              hipStream_t stream)
{
    const float* img    = (const float*)d_in[0];
    const float* hidden = (const float*)d_in[1];
    const float* Wq     = (const float*)d_in[2];
    const float* bq     = (const float*)d_in[3];
    const float* Wk     = (const float*)d_in[4];
    const float* bk     = (const float*)d_in[5];
    const float* Wv     = (const float*)d_in[6];
    const float* bv     = (const float*)d_in[7];

    float* ws    = (float*)d_ws;
    float* Qh    = ws;                          /* 256*512 floats  */
    float* e_ws  = ws + (size_t)BATCH * HIDN;   /* 50176 floats    */

    float* ctx   = (float*)d_out;               /* 256*2048 floats */
    float* alpha = ctx + (size_t)BATCH * EMB;   /* 256*196 floats  */

    qh_kernel<<<BATCH, 256, 0, stream>>>(hidden, Wq, bq, Qh);
    att_gemm_kernel<<<NROW / 64, 256, 0, stream>>>(img, Wk, bk, Qh, Wv, bv, e_ws);
    softmax_kernel<<<BATCH, 256, 0, stream>>>(e_ws, alpha);
    context_kernel<<<BATCH * 2, 256, 0, stream>>>(img, alpha, ctx);
}